// MMDLoss_40561671143496
// MI455X (gfx1250) — compile-verified
//
#include <hip/hip_runtime.h>

// ---------------------------------------------------------------------------
// MMD loss with farthest-neighbor selection, MI455X (gfx1250).
//
// Roofline: selection streams 107 MB from HBM (~4.6us @ 23.3 TB/s).
// GEMM phase: 3 x (2048x2048x768) f32 Grams = 19.3 GFLOP, with K_xx/K_yy
// computed lower-triangle-only (exact by symmetry) -> ~13 GF effective.
// X/Y (12.6 MB) live in the 192 MB L2, so this phase is matrix-pipe bound:
//   V_WMMA_F32_16X16X4_F32  (D[16x16] += A[16x4] * B[4x16], wave32)
// Each wave owns a 32x32 macro-tile: 4 independent accumulator chains hide
// the XDL accumulation latency, and each loaded fragment feeds 2 WMMAs
// (1 load : 1 wmma instead of 2 : 1 -> 8 FLOP/B from L2).
// ---------------------------------------------------------------------------

#define UPSAMPLE 4
#define GAMMA    0.00065f
#define C_DIM    768
#define B_DIM    2
#define H1       32
#define W1       32
#define N_PTS    (B_DIM * H1 * W1)      // 2048
#define H2       (H1 * UPSAMPLE)        // 128
#define W2       (W1 * UPSAMPLE)        // 128

typedef __attribute__((ext_vector_type(2))) float v2f;
typedef __attribute__((ext_vector_type(8))) float v8f;

// ---------------------------------------------------------------------------
// Phase 1: per-site farthest-of-16 selection + pack X, Y_sel row-major (N x C)
// One wave per (b, h) row; lane = w. HBM-streaming bound; y's 4x4 block is
// fully consumed by the dh/dw unroll so every fetched line is used.
// ---------------------------------------------------------------------------
__global__ void __launch_bounds__(32)
select_kernel(const float* __restrict__ x, const float* __restrict__ y,
              float* __restrict__ Xm, float* __restrict__ Ym,
              float* __restrict__ xn, float* __restrict__ yn,
              float* __restrict__ acc) {
    const int lane = threadIdx.x;           // w coordinate
    const int b = blockIdx.x / H1;
    const int h = blockIdx.x % H1;
    const int w = lane;
    const int n = (b * H1 + h) * W1 + w;

    // re-zero the global accumulator every call (ws is not re-poisoned).
    if (blockIdx.x == 0 && lane == 0) acc[0] = 0.0f;

    float d2[16];
#pragma unroll
    for (int j = 0; j < 16; ++j) d2[j] = 0.0f;

    float xsum = 0.0f;
    for (int c = 0; c < C_DIM; ++c) {
        const float xv = x[((size_t)(b * C_DIM + c) * H1 + h) * W1 + w];
        xsum += xv * xv;
        const float* yc = y + ((size_t)(b * C_DIM + c) * H2 + h * UPSAMPLE) * W2
                            + w * UPSAMPLE;
#pragma unroll
        for (int dh = 0; dh < UPSAMPLE; ++dh) {
#pragma unroll
            for (int dw = 0; dw < UPSAMPLE; ++dw) {
                const float yv = yc[dh * W2 + dw];
                const float df = xv - yv;
                d2[dh * UPSAMPLE + dw] += df * df;
            }
        }
    }

    // argmax with first-occurrence tie-break (matches jnp.argmax)
    int best = 0;
    float bv = d2[0];
#pragma unroll
    for (int j = 1; j < 16; ++j) {
        if (d2[j] > bv) { bv = d2[j]; best = j; }
    }
    const int dh = best >> 2;
    const int dw = best & 3;

    // second streaming pass: pack rows + row norms (x re-read hits caches)
    float ysum = 0.0f;
    for (int c = 0; c < C_DIM; ++c) {
        const float xv = x[((size_t)(b * C_DIM + c) * H1 + h) * W1 + w];
        const float yv = y[((size_t)(b * C_DIM + c) * H2 + h * UPSAMPLE + dh) * W2
                           + w * UPSAMPLE + dw];
        Xm[n * C_DIM + c] = xv;
        Ym[n * C_DIM + c] = yv;
        ysum += yv * yv;
    }
    xn[n] = xsum;
    yn[n] = ysum;
}

// ---------------------------------------------------------------------------
// Phase 2: one wave per 32x32 macro-tile (2x2 WMMA tiles), blockIdx.z selects
// XX / YY / XY. For the symmetric XX/YY Grams we compute only macro-tiles with
// bx <= by and double the strictly-lower ones (bitwise-exact by symmetry).
//
// f32 WMMA fragment layouts (ISA 7.12.2, 32-bit A 16x4 / C 16x16):
//   A: lane l holds row M=l%16; VGPR pair = K = {k+2*(l/16), k+2*(l/16)+1}
//   B: mirrored (one contiguous float2 per lane from the transposed operand)
//   C/D: vgpr r, lane l -> M = r + 8*(l/16), N = l%16
// The early-out is workgroup-uniform, so every wave reaching the WMMA loop
// has EXEC all-ones as the ISA requires.
// ---------------------------------------------------------------------------
__global__ void __launch_bounds__(32)
gram_kernel(const float* __restrict__ Xm, const float* __restrict__ Ym,
            const float* __restrict__ xn, const float* __restrict__ yn,
            float* __restrict__ acc) {
    const int mode = blockIdx.z;            // 0: XX, 1: YY, 2: XY
    const int bx = blockIdx.x;              // macro row-tile (M)
    const int by = blockIdx.y;              // macro col-tile (N)

    // symmetric modes: skip upper triangle of macro-tiles (uniform exit)
    if (mode != 2 && bx > by) return;

    const float* Am = (mode == 1) ? Ym : Xm;
    const float* Bm = (mode == 0) ? Xm : Ym;
    const float* an = (mode == 1) ? yn : xn;
    const float* bn = (mode == 0) ? xn : yn;
    float wgt = (mode == 2) ? -2.0f : 1.0f;
    if (mode != 2 && bx < by) wgt *= 2.0f;  // count the mirrored tile

    const int lane = threadIdx.x;
    const int half = lane >> 4;             // 0: K lo pair, 1: K hi pair
    const int l16  = lane & 15;

    const float* ap0 = Am + (size_t)(bx * 32 + l16) * C_DIM + 2 * half;
    const float* ap1 = ap0 + 16 * C_DIM;
    const float* bp0 = Bm + (size_t)(by * 32 + l16) * C_DIM + 2 * half;
    const float* bp1 = bp0 + 16 * C_DIM;

    v8f c00 = {}, c01 = {}, c10 = {}, c11 = {};
    for (int k = 0; k < C_DIM; k += 4) {
        const v2f a0 = *(const v2f*)(ap0 + k);  // 8B aligned (k even)
        const v2f a1 = *(const v2f*)(ap1 + k);
        const v2f b0 = *(const v2f*)(bp0 + k);
        const v2f b1 = *(const v2f*)(bp1 + k);
        // 8 args: (neg_a, A, neg_b, B, c_mod, C, reuse_a, reuse_b)
        c00 = __builtin_amdgcn_wmma_f32_16x16x4_f32(false, a0, false, b0,
                                                    (short)0, c00, false, false);
        c01 = __builtin_amdgcn_wmma_f32_16x16x4_f32(false, a0, false, b1,
                                                    (short)0, c01, false, false);
        c10 = __builtin_amdgcn_wmma_f32_16x16x4_f32(false, a1, false, b0,
                                                    (short)0, c10, false, false);
        c11 = __builtin_amdgcn_wmma_f32_16x16x4_f32(false, a1, false, b1,
                                                    (short)0, c11, false, false);
    }

    // epilogue: d = |a|^2 + |b|^2 - 2<a,b>, clamp, exp, macro-tile sum
    const float bn0 = bn[by * 32 + l16];
    const float bn1 = bn[by * 32 + 16 + l16];
    float s = 0.0f;
#pragma unroll
    for (int r = 0; r < 8; ++r) {
        const float an0 = an[bx * 32 + r + 8 * half];
        const float an1 = an[bx * 32 + 16 + r + 8 * half];
        float d;
        d = fmaxf(an0 + bn0 - 2.0f * c00[r], 0.0f); s += __expf(-GAMMA * d);
        d = fmaxf(an0 + bn1 - 2.0f * c01[r], 0.0f); s += __expf(-GAMMA * d);
        d = fmaxf(an1 + bn0 - 2.0f * c10[r], 0.0f); s += __expf(-GAMMA * d);
        d = fmaxf(an1 + bn1 - 2.0f * c11[r], 0.0f); s += __expf(-GAMMA * d);
    }
#pragma unroll
    for (int off = 16; off > 0; off >>= 1) s += __shfl_xor(s, off, 32);
    if (lane == 0) atomicAdd(acc, wgt * s);
}

__global__ void finish_kernel(const float* __restrict__ acc,
                              float* __restrict__ out) {
    out[0] = acc[0] * (1.0f / ((float)N_PTS * (float)N_PTS));
}

extern "C" void kernel_launch(void* const* d_in, const int* in_sizes, int n_in,
                              void* d_out, int out_size, void* d_ws, size_t ws_size,
                              hipStream_t stream) {
    const float* x = (const float*)d_in[0];   // (2, 768, 32, 32)
    const float* y = (const float*)d_in[1];   // (2, 768, 128, 128)
    float* out = (float*)d_out;               // scalar

    // workspace layout (~12.6 MB): X | Y | xn | yn | acc
    float* Xm  = (float*)d_ws;
    float* Ym  = Xm + (size_t)N_PTS * C_DIM;
    float* xn  = Ym + (size_t)N_PTS * C_DIM;
    float* yn  = xn + N_PTS;
    float* acc = yn + N_PTS;

    select_kernel<<<B_DIM * H1, 32, 0, stream>>>(x, y, Xm, Ym, xn, yn, acc);
    gram_kernel<<<dim3(N_PTS / 32, N_PTS / 32, 3), 32, 0, stream>>>(Xm, Ym, xn, yn, acc);
    finish_kernel<<<1, 1, 0, stream>>>(acc, out);
}